// GeluAvgEmbed_87823491269195
// MI455X (gfx1250) — compile-verified
//
#include <hip/hip_runtime.h>
#include <math.h>

// CDNA5 / gfx1250: wave32, WMMA 16x16 shapes.
typedef __attribute__((ext_vector_type(2))) float v2f;
typedef __attribute__((ext_vector_type(8))) float v8f;

#define CELLS_PER_BLOCK 16
#define THREADS 512          // 16 wave32 waves; one cell per wave
#define LDS_PITCH 132        // 128 + 4 pad -> conflict-free strided column reads
#define TOK 32
#define DIM 128

__device__ __forceinline__ float gelu_erf(float v) {
    // exact GELU: 0.5*x*(1+erf(x/sqrt(2))) — matches jax approximate=False
    return 0.5f * v * (1.0f + erff(v * 0.70710678118654752440f));
}

__global__ __launch_bounds__(THREADS) void GeluAvgEmbed_kernel(
    const int*   __restrict__ x,     // [B,R,C,T] flat = [ncells, 32]
    const float* __restrict__ W,     // [32000, 128]
    const float* __restrict__ wp,    // [1, 128]
    const float* __restrict__ bp,    // [1]
    float*       __restrict__ out)   // [ncells]
{
    __shared__ __align__(16) float lds[CELLS_PER_BLOCK * LDS_PITCH];

    const int lane  = threadIdx.x & 31;
    const int wave  = threadIdx.x >> 5;
    const int cell0 = blockIdx.x * CELLS_PER_BLOCK;
    const int cell  = cell0 + wave;

    // ---- Stage 1: gather + mean-pool + GELU (one wave per cell) ----
    // lane t holds token t's index; broadcast via v_readlane so the row
    // base address is scalar (SGPR) -> global_load_b128 with SADDR.
    const int* xc = x + (size_t)cell * TOK;
    int myidx = xc[lane];

    float ax = 0.f, ay = 0.f, az = 0.f, aw = 0.f;
#pragma unroll
    for (int t = 0; t < TOK; ++t) {
        int idx = __builtin_amdgcn_readlane(myidx, t);     // uniform token id
        const float4* row = (const float4*)(W + (size_t)idx * DIM);
        float4 v = row[lane];                              // lane l: dims 4l..4l+3
        ax += v.x; ay += v.y; az += v.z; aw += v.w;
    }
    const float inv = 1.0f / (float)TOK;
    float4 actv;
    actv.x = gelu_erf(ax * inv);
    actv.y = gelu_erf(ay * inv);
    actv.z = gelu_erf(az * inv);
    actv.w = gelu_erf(aw * inv);

    // Stage activations: row = cell-in-block (pitch 132 f32), cols = dims.
    // byte addr = wave*528 + lane*16 : both multiples of 16 -> aligned b128.
    *(float4*)&lds[wave * LDS_PITCH + lane * 4] = actv;

    __syncthreads();

    // ---- Stage 2: 16-cell dot product via V_WMMA_F32_16X16X4_F32 ----
    // D[m,n] = sum_k A[m,k]*B[k,n], accumulated over 32 chunks of K=4.
    // A layout (32-bit 16x4): lanes 0-15 = rows m, VGPR0/1 = K0/K1;
    //                         lanes 16-31 = rows m, VGPR0/1 = K2/K3.
    // B columns all equal w -> value depends only on K.
    if (wave == 0) {
        const int half = lane >> 4;       // 0: K{0,1}, 1: K{2,3}
        const int off  = half << 1;
        const int m    = lane & 15;

        v8f c = {};
#pragma unroll
        for (int j = 0; j < DIM / 4; ++j) {
            v2f a = *(const v2f*)&lds[m * LDS_PITCH + 4 * j + off];
            v2f b;
            b.x = wp[4 * j + off];
            b.y = wp[4 * j + off + 1];
            // (neg_a, A, neg_b, B, c_mod, C, reuse_a, reuse_b)
            c = __builtin_amdgcn_wmma_f32_16x16x4_f32(
                    false, a, false, b, (short)0, c, false, false);
        }

        // C/D layout: element r, lanes 0-15 -> cell r; lanes 16-31 -> cell r+8.
        if (m == 0) {
            const float bias = bp[0];
            float* o = out + cell0 + half * 8;
#pragma unroll
            for (int r = 0; r < 8; ++r)
                o[r] = c[r] + bias;
        }
    }
}

extern "C" void kernel_launch(void* const* d_in, const int* in_sizes, int n_in,
                              void* d_out, int out_size, void* d_ws, size_t ws_size,
                              hipStream_t stream) {
    const int*   x  = (const int*)d_in[0];    // [8,100,64,32] int32
    const float* W  = (const float*)d_in[1];  // [32000,128] f32
    const float* wp = (const float*)d_in[2];  // [1,128] f32
    const float* bp = (const float*)d_in[3];  // [1] f32
    float* out = (float*)d_out;               // [8,100,64] f32 = 51200

    const int ncells = out_size;              // 51200, multiple of 16
    const int blocks = ncells / CELLS_PER_BLOCK;
    GeluAvgEmbed_kernel<<<blocks, THREADS, 0, stream>>>(x, W, wp, bp, out);
}